// RecurrentForgettingAttentionCell_74775380623498
// MI455X (gfx1250) — compile-verified
//
#include <hip/hip_runtime.h>
#include <hip/hip_bf16.h>
#include <stdint.h>

#define BATCH  4096
#define D      128
#define TILE_K 16
#define NTILES (D / TILE_K)   // 8
#define EPS_F  1e-6f

typedef __attribute__((ext_vector_type(2))) float v2f;
typedef __attribute__((ext_vector_type(8))) float v8f;

// One workgroup (128 threads = 4 wave32) per batch row b.
// Wave w owns v-columns [32w, 32w+32) as two 16x16 WMMA tiles per k-step.
__global__ __launch_bounds__(128) void rfa_cell_kernel(
    const float* __restrict__ q_g,  const float* __restrict__ k_g,
    const float* __restrict__ v_g,  const float* __restrict__ S_g,
    const float* __restrict__ z_g,  const float* __restrict__ wq_g,
    const float* __restrict__ wk_g, const float* __restrict__ bias_g,
    float* __restrict__ out_g, float* __restrict__ Snew_g,
    float* __restrict__ znew_g, float* __restrict__ f_g)
{
    __shared__ __align__(16) float sbuf[2][TILE_K * D];  // 2 x 8 KB double buffer
    __shared__ float qs[D];
    __shared__ float ks[D];
    __shared__ float vs[D];
    __shared__ float red[D];

    const int b   = blockIdx.x;
    const int tid = threadIdx.x;
    const size_t bD = (size_t)b * D;

    const float qv = q_g[bD + tid];
    const float kv = k_g[bD + tid];
    const float vv = v_g[bD + tid];
    const float zv = z_g[bD + tid];
    qs[tid] = qv;
    ks[tid] = kv;
    vs[tid] = vv;

    const float* Sb  = S_g    + (size_t)b * D * D;
    float*       Snb = Snew_g + (size_t)b * D * D;
    const uint64_t sb64 = (uint64_t)(uintptr_t)Sb;

    // Async DMA of one 16x128 fp32 tile of S[b] into LDS (ASYNCcnt path).
    // 4 instructions/tile/wave * 4 waves * 32 lanes * 16B = 8192 B = full tile.
    auto issue_tile = [&](int t) {
        const uint32_t lane16   = (uint32_t)(tid * 16);
        const uint32_t lds_base = (uint32_t)(uintptr_t)(&sbuf[t & 1][0]);
        #pragma unroll
        for (int j = 0; j < 4; ++j) {
            const uint32_t goff  = (uint32_t)(t * (TILE_K * D * 4) + j * 2048) + lane16;
            const uint32_t laddr = lds_base + (uint32_t)(j * 2048) + lane16;
            asm volatile("global_load_async_to_lds_b128 %0, %1, %2"
                         :: "v"(laddr), "v"(goff), "s"(sb64)
                         : "memory");
        }
    };

    // Kick off the first two tiles; DMA overlaps the gate math below.
    issue_tile(0);
    issue_tile(1);

    // f = sigmoid(q.w_q + k.w_k + bias)  (128-wide LDS tree reduction)
    red[tid] = qv * wq_g[tid] + kv * wk_g[tid];
    __syncthreads();
    #pragma unroll
    for (int s = 64; s >= 1; s >>= 1) {
        if (tid < s) red[tid] += red[tid + s];
        __syncthreads();
    }
    const float f = 1.0f / (1.0f + __expf(-(red[0] + bias_g[0])));
    __syncthreads();

    // z_new = f*z + k ; den = q.z_new + eps
    const float zn = f * zv + kv;
    znew_g[bD + tid] = zn;
    red[tid] = qv * zn;
    __syncthreads();
    #pragma unroll
    for (int s = 64; s >= 1; s >>= 1) {
        if (tid < s) red[tid] += red[tid + s];
        __syncthreads();
    }
    const float inv_den = 1.0f / (red[0] + EPS_F);

    // WMMA lane mapping (wave32): lane = h*16 + n ; D-tile VGPR r holds
    // rows (r + 8h) across columns n.  Wave w owns v-blocks 32w and 32w+16.
    const int wave = tid >> 5;
    const int lane = tid & 31;
    const int h    = lane >> 4;    // half-wave select
    const int n    = lane & 15;    // column within 16-wide tile
    const int vb0  = wave * 32;
    const int vb1  = vb0 + 16;

    float num0 = 0.0f, num1 = 0.0f;

    for (int t = 0; t < NTILES; ++t) {
        // Each wave issued 4 async loads per tile; steady state has tiles
        // t and t+1 outstanding (8).  <=4 guarantees tile t landed in LDS.
        if (t < NTILES - 1) {
            asm volatile("s_wait_asynccnt 0x4" ::: "memory");
        } else {
            asm volatile("s_wait_asynccnt 0x0" ::: "memory");
        }
        __syncthreads();   // all 4 waves' DMA chunks visible

        const float* sl = &sbuf[t & 1][0];
        const int kblk = t * TILE_K;

        // Rank-1 operands: only the K=0 slice is populated.
        // A (16x4): lane m<16, VGPR0 = k_b[kblk+m]; all other slots zero.
        // B (4x16): lane n<16, VGPR0 = v_b[vblk+n]; all other slots zero.
        v2f a; a.x = (h == 0) ? ks[kblk + n] : 0.0f; a.y = 0.0f;

        #pragma unroll
        for (int tile = 0; tile < 2; ++tile) {
            const int vb = tile ? vb1 : vb0;
            v2f bm; bm.x = (h == 0) ? vs[vb + n] : 0.0f; bm.y = 0.0f;

            // C = f * S_tile in the 16x16 f32 C/D layout
            v8f c;
            #pragma unroll
            for (int r = 0; r < 8; ++r)
                c[r] = f * sl[(r + 8 * h) * D + vb + n];

            // S_new_tile = A x B + C  ==  f*S + k v^T   (exact fp32 FMA)
            v8f d = __builtin_amdgcn_wmma_f32_16x16x4_f32(
                false, a, false, bm, (short)0, c, false, false);

            float acc = 0.0f;
            #pragma unroll
            for (int r = 0; r < 8; ++r) {
                const int krow = kblk + r + 8 * h;
                // 256MB streamed once: NT store, two 64B segments per row-pair
                __builtin_nontemporal_store(d[r], &Snb[(size_t)krow * D + vb + n]);
                acc = __builtin_fmaf(qs[krow], d[r], acc);
            }
            if (tile) num1 += acc; else num0 += acc;
        }

        __syncthreads();   // everyone done reading this buffer
        if (t + 2 < NTILES) issue_tile(t + 2);   // refill the freed buffer
    }

    // Lanes l and l^16 hold row-halves of the same column: fold them.
    num0 += __shfl_xor(num0, 16, 32);
    num1 += __shfl_xor(num1, 16, 32);

    const int   col = vb0 + h * 16 + n;          // h=0 -> tile0 col, h=1 -> tile1 col
    const float res = (h == 0) ? num0 : num1;
    out_g[bD + col] = res * inv_den;

    if (tid == 0) f_g[b] = f;
}

extern "C" void kernel_launch(void* const* d_in, const int* in_sizes, int n_in,
                              void* d_out, int out_size, void* d_ws, size_t ws_size,
                              hipStream_t stream) {
    (void)in_sizes; (void)n_in; (void)out_size; (void)d_ws; (void)ws_size;

    const float* q    = (const float*)d_in[0];
    const float* k    = (const float*)d_in[1];
    const float* v    = (const float*)d_in[2];
    const float* S    = (const float*)d_in[3];
    const float* z    = (const float*)d_in[4];
    const float* wq   = (const float*)d_in[5];
    const float* wk   = (const float*)d_in[6];
    const float* bias = (const float*)d_in[7];

    // d_out = [output (B,D)] [S_new (B,D,D)] [z_new (B,D)] [f (B,1)]
    float* out  = (float*)d_out;
    float* Snew = out  + (size_t)BATCH * D;
    float* znew = Snew + (size_t)BATCH * D * D;
    float* fout = znew + (size_t)BATCH * D;

    rfa_cell_kernel<<<BATCH, D, 0, stream>>>(q, k, v, S, z, wq, wk, bias,
                                             out, Snew, znew, fout);
}